// CVADecoder_21698174780139
// MI455X (gfx1250) — compile-verified
//
#include <hip/hip_runtime.h>
#include <cstdint>

#define BATCH      512
#define T_TOTAL    3072
#define CLIP_VAL   20.0f
// generator masks over reg = [b, p5..p0] (bit6 = b):
// g0 = 1111001b = 0x79, g1 = 1011011b = 0x5B
#define G0_MASK 0x79
#define G1_MASK 0x5B

typedef float v2f __attribute__((ext_vector_type(2)));
typedef float v8f __attribute__((ext_vector_type(8)));

#if __has_builtin(__builtin_amdgcn_global_load_async_to_lds_b128)
#define CVA_ASYNC_STAGE 1
typedef int v4i_t __attribute__((ext_vector_type(4)));
typedef __attribute__((address_space(1))) v4i_t as1_v4i;
typedef __attribute__((address_space(3))) v4i_t as3_v4i;
#else
#define CVA_ASYNC_STAGE 0
#endif

__device__ __forceinline__ void edge_weights(int s, int j,
                                             float& w0, float& w1, float& wp) {
  int prev = (2 * s + j) & 63;
  int w7   = ((s >> 5) << 6) | prev;          // [b, p5..p0]
  int c0   = __popc(w7 & G0_MASK) & 1;
  int c1   = __popc(w7 & G1_MASK) & 1;
  if (c0 & c1) { w0 = 0.f; w1 = 0.f; wp = 1.f; }   // use WMMA pair-sum
  else         { w0 = (float)c0; w1 = (float)c1; wp = 0.f; }
}

// One wave per batch row. 4 waves / block (48 KB LDS).
__global__ __launch_bounds__(128)
void cva_acs_kernel(const float* __restrict__ x,
                    unsigned long long* __restrict__ dec) {
  __shared__ __align__(16) float sx[4][2048];   // staged LLR row per wave
  __shared__ __align__(16) float sps[4][1024];  // WMMA pair sums per wave

  const int warp = threadIdx.x >> 5;
  const int lane = threadIdx.x & 31;
  const int b    = blockIdx.x * 4 + warp;

  // ---- stage this wave's LLR row into LDS ----
#if CVA_ASYNC_STAGE
  {
    const float* gsrc = x + (size_t)b * 2048;
    float*       ldst = sx[warp];
#pragma unroll
    for (int k = 0; k < 16; ++k) {
      const int e = (k * 32 + lane) * 4;   // float index; 16B per lane, coalesced
      __builtin_amdgcn_global_load_async_to_lds_b128(
          (as1_v4i*)(uintptr_t)(gsrc + e),
          (as3_v4i*)(uint32_t)(uintptr_t)(ldst + e),
          0, 0);
    }
#if __has_builtin(__builtin_amdgcn_s_wait_asynccnt)
    __builtin_amdgcn_s_wait_asynccnt(0);
#else
    asm volatile("s_wait_asynccnt 0x0" ::: "memory");
#endif
  }
#else
  {
    const float4* g4 = (const float4*)(x + (size_t)b * 2048);
    float4*       s4 = (float4*)sx[warp];
#pragma unroll
    for (int k = 0; k < 16; ++k) s4[k * 32 + lane] = g4[k * 32 + lane];
  }
#endif

  // ---- WMMA: psum[i] = x[2i] + x[2i+1] for i = 0..1023 ----
  // A (16x4 f32): lanes 0-15 carry rows M=0..15 with K0=llr0, K1=llr1;
  // lanes 16-31 (K2/K3) zeroed via cndmask. B = all-ones, so D[m][*] =
  // llr0+llr1 regardless of K sub-layout; f32 RNE = bit-exact vs VALU add.
  {
    float* sxr = sx[warp];
    float* psr = sps[warp];
    for (int g = 0; g < 64; ++g) {
      const int i = g * 16 + (lane & 15);
      float a0 = sxr[2 * i];
      float a1 = sxr[2 * i + 1];
      if (lane >= 16) { a0 = 0.f; a1 = 0.f; }
      v2f A;  A.x = a0;  A.y = a1;
      v2f Bm; Bm.x = 1.f; Bm.y = 1.f;
      v8f C = {0.f, 0.f, 0.f, 0.f, 0.f, 0.f, 0.f, 0.f};
      v8f D = __builtin_amdgcn_wmma_f32_16x16x4_f32(false, A, false, Bm,
                                                    (short)0, C, false, false);
      // C/D layout: VGPR r = row r (lanes 0-15) / row r+8 (lanes 16-31)
      if (lane == 0) {
#pragma unroll
        for (int r = 0; r < 8; ++r) psr[g * 16 + r] = D[r];
      } else if (lane == 16) {
#pragma unroll
        for (int r = 0; r < 8; ++r) psr[g * 16 + 8 + r] = D[r];
      }
    }
  }

  // ---- per-lane constant edge weights: lane l owns new states l and l+32,
  //      whose predecessors are exactly the pair (prob[2l], prob[2l+1]) ----
  float aw0, aw1, awp, bw0, bw1, bwp, cw0, cw1, cwp, dw0, dw1, dwp;
  edge_weights(lane,      0, aw0, aw1, awp);
  edge_weights(lane,      1, bw0, bw1, bwp);
  edge_weights(lane + 32, 0, cw0, cw1, cwp);
  edge_weights(lane + 32, 1, dw0, dw1, dwp);

  const int  idx0  = (2 * lane) & 31;       // re-pack bpermute sources
  const int  idx1  = (2 * lane + 1) & 31;
  const bool isLow = lane < 16;

  const float2* sx2 = (const float2*)sx[warp];
  const float*  psr = sps[warp];

  float p0 = 0.f, p1 = 0.f;                 // pair (prob[2l], prob[2l+1])

  // one trellis step: ACS for states (lane, lane+32) + pair re-pack
  auto step = [&](int t, bool& iLo, bool& iHi) {
    const int    col = t & 1023;            // llr_t = x[(2t)%2048 .. +1]
    const float2 llr = sx2[col];            // wave-uniform -> LDS broadcast
    const float  ps  = psr[col];

    const float brA = fmaf(aw0, llr.x, fmaf(aw1, llr.y, awp * ps));
    const float brB = fmaf(bw0, llr.x, fmaf(bw1, llr.y, bwp * ps));
    const float brC = fmaf(cw0, llr.x, fmaf(cw1, llr.y, cwp * ps));
    const float brD = fmaf(dw0, llr.x, fmaf(dw1, llr.y, dwp * ps));

    const float t0 = p0 + brA, t1 = p1 + brB;
    iLo = t1 < t0;                          // argmin picks j=1 only on strict <
    float nLo = fminf(t0, t1);
    nLo = fminf(fmaxf(nLo, -CLIP_VAL), CLIP_VAL);

    const float u0 = p0 + brC, u1 = p1 + brD;
    iHi = u1 < u0;
    float nHi = fminf(u0, u1);
    nHi = fminf(fmaxf(nHi, -CLIP_VAL), CLIP_VAL);

    // re-pack: next pair (new[2l], new[2l+1]); slot picked by dest half
    const float g00 = __shfl(nLo, idx0, 32);
    const float g01 = __shfl(nHi, idx0, 32);
    const float g10 = __shfl(nLo, idx1, 32);
    const float g11 = __shfl(nHi, idx1, 32);
    p0 = isLow ? g00 : g01;
    p1 = isLow ? g10 : g11;
  };

  // phase 1: t in [0,1024) — decisions never read by traceback, no ballots
  for (int t = 0; t < 1024; ++t) {
    bool iLo, iHi;
    step(t, iLo, iHi);
  }

  // phase 2: t in [1024,3072) — capture uniform ballot word into lane t&31,
  // flush 32 steps at a time as one coalesced 256B global_store_b64
  unsigned long long* drow = dec + (size_t)b * 2048;
  for (int tb = 1024; tb < T_TOTAL; tb += 32) {
    unsigned long long myword = 0;
    for (int k = 0; k < 32; ++k) {
      bool iLo, iHi;
      step(tb + k, iLo, iHi);
      const unsigned mLo = __builtin_amdgcn_ballot_w32(iLo);
      const unsigned mHi = __builtin_amdgcn_ballot_w32(iHi);
      const unsigned long long w =
          ((unsigned long long)mHi << 32) | (unsigned long long)mLo;
      if (lane == k) myword = w;            // uniform -> v_cndmask capture
    }
    drow[(tb - 1024) + lane] = myword;
  }
}

// One thread per batch row; private sequential u64 stream per thread.
__global__ __launch_bounds__(128)
void cva_traceback_kernel(const unsigned long long* __restrict__ dec,
                          float* __restrict__ out) {
  const int b = blockIdx.x * blockDim.x + threadIdx.x;
  if (b >= BATCH) return;
  const unsigned long long* row = dec + (size_t)b * 2048;
  int state = 0;  // traceback starts at state 0 after step T-1
  // t in [2048,3072): state updates only (bits discarded)
  for (int t = T_TOTAL - 1; t >= 2048; --t) {
    const unsigned long long w = row[t - 1024];
    state = ((state << 1) | (int)((w >> state) & 1ull)) & 63;
  }
  // t in [1024,2048): emit bit (before update), then update
  for (int t = 2047; t >= 1024; --t) {
    out[(size_t)b * 1024 + (t - 1024)] = (float)((state + 1) & 1);
    const unsigned long long w = row[t - 1024];
    state = ((state << 1) | (int)((w >> state) & 1ull)) & 63;
  }
}

extern "C" void kernel_launch(void* const* d_in, const int* in_sizes, int n_in,
                              void* d_out, int out_size, void* d_ws, size_t ws_size,
                              hipStream_t stream) {
  (void)in_sizes; (void)n_in; (void)out_size; (void)ws_size;
  const float* x = (const float*)d_in[0];
  float* out = (float*)d_out;
  unsigned long long* dec = (unsigned long long*)d_ws;  // 512*2048*8 = 8.4 MB

  cva_acs_kernel<<<BATCH / 4, 128, 0, stream>>>(x, dec);
  cva_traceback_kernel<<<(BATCH + 127) / 128, 128, 0, stream>>>(dec, out);
}